// CIN_86380382257180
// MI455X (gfx1250) — compile-verified
//
#include <hip/hip_runtime.h>

typedef float v2f __attribute__((ext_vector_type(2)));
typedef float v4f __attribute__((ext_vector_type(4)));
typedef float v8f __attribute__((ext_vector_type(8)));

#define WAVES_PER_BLOCK 4
#define NB 2
#define XK_STRIDE 66   // 64 + 2 pad -> conflict-free ds_load_b64 / ds_store_b32
#define X0_STRIDE 34   // 32 + 2 pad

// packed-W region sizes in v4 (16B) entries: 32*HK1*64/4 entries per layer
#define L0_V4 16384    // HK1=32
#define L1_V4 32768    // HK1=64
#define PACK_V4_TOTAL (L0_V4 + 2 * L1_V4)   // 81920 entries = 1.31 MB

// D = A(16x4 f32) * B(4x16 f32) + C(16x16 f32), wave32 WMMA
__device__ __forceinline__ v8f wmma4(v2f a, v2f b, v8f c) {
  return __builtin_amdgcn_wmma_f32_16x16x4_f32(
      /*neg_a=*/false, a, /*neg_b=*/false, b,
      /*c_mod=*/(short)0, c, /*reuse_a=*/false, /*reuse_b=*/false);
}

// ---------------- packed hot loop: 1 global_load_b128 per 4 WMMAs ----------
// Packed layout: entry((m*(HK1/4)+g)*2+np, lane) -> 4 floats:
//   { W[m,4g+kk,np*32+col], W[m,4g+kk+1,np*32+col],
//     W[m,4g+kk,np*32+col+16], W[m,4g+kk+1,np*32+col+16] }
template <int HK1>
__device__ __forceinline__ void cross_layer_packed(
    const float* __restrict__ Wp, const float* __restrict__ s0,
    const float* __restrict__ s1, const v2f (&xkf)[NB][HK1 / 4],
    v8f (&acc)[NB][4], int lane) {
  const float* wp = Wp + lane * 4;
  for (int m = 0; m < 32; ++m) {
    float sm0 = s0[m];  // ds_load_b32: per-lane scalar x0[d,m]
    float sm1 = s1[m];
#pragma unroll
    for (int g = 0; g < HK1 / 4; ++g) {
      v2f a0, a1;
      a0.x = xkf[0][g].x * sm0; a0.y = xkf[0][g].y * sm0;
      a1.x = xkf[1][g].x * sm1; a1.y = xkf[1][g].y * sm1;
#pragma unroll
      for (int np = 0; np < 2; ++np) {
        v4f w4 = *(const v4f*)(wp + g * 256 + np * 128);  // b128, 512B/wave
        v2f wa, wb;
        wa.x = w4.x; wa.y = w4.y;   // B-frag for nt = 2*np
        wb.x = w4.z; wb.y = w4.w;   // B-frag for nt = 2*np+1
        acc[0][2 * np]     = wmma4(a0, wa, acc[0][2 * np]);
        acc[1][2 * np]     = wmma4(a1, wa, acc[1][2 * np]);
        acc[0][2 * np + 1] = wmma4(a0, wb, acc[0][2 * np + 1]);
        acc[1][2 * np + 1] = wmma4(a1, wb, acc[1][2 * np + 1]);
      }
    }
    wp += HK1 * 64;
  }
}

// ---------------- fallback (unpacked W), used only if ws too small ---------
template <int HK1>
__device__ __forceinline__ void cross_layer_plain(
    const float* __restrict__ Wl, const float* __restrict__ s0,
    const float* __restrict__ s1, const v2f (&xkf)[NB][HK1 / 4],
    v8f (&acc)[NB][4], int koff) {
  const float* wp = Wl + koff;
  for (int m = 0; m < 32; ++m) {
    float sm0 = s0[m];
    float sm1 = s1[m];
#pragma unroll
    for (int g = 0; g < HK1 / 4; ++g) {
      v2f a0, a1;
      a0.x = xkf[0][g].x * sm0; a0.y = xkf[0][g].y * sm0;
      a1.x = xkf[1][g].x * sm1; a1.y = xkf[1][g].y * sm1;
#pragma unroll
      for (int nt = 0; nt < 4; ++nt) {
        v2f w;
        w.x = wp[g * 256 + nt * 16];
        w.y = wp[g * 256 + nt * 16 + 64];
        acc[0][nt] = wmma4(a0, w, acc[0][nt]);
        acc[1][nt] = wmma4(a1, w, acc[1][nt]);
      }
    }
    wp += HK1 * 64;
  }
}

template <int HK1, bool PACKED>
__device__ __forceinline__ void cross_layer(
    const float* __restrict__ Wl, const float* __restrict__ s0,
    const float* __restrict__ s1, const v2f (&xkf)[NB][HK1 / 4],
    v8f (&acc)[NB][4], int lane, int koff) {
  if (PACKED)
    cross_layer_packed<HK1>(Wl, s0, s1, xkf, acc, lane);
  else
    cross_layer_plain<HK1>(Wl, s0, s1, xkf, acc, koff);
}

__device__ __forceinline__ void store_sums(const v8f (&acc)[NB][4],
                                           float* __restrict__ out, long b0,
                                           int layer, int lane) {
#pragma unroll
  for (int n = 0; n < NB; ++n) {
    float p[4];
#pragma unroll
    for (int nt = 0; nt < 4; ++nt) {
      float t = 0.0f;
#pragma unroll
      for (int j = 0; j < 8; ++j) t += acc[n][nt][j];  // sum over d (half)
      t += __shfl_xor(t, 16, 32);                      // combine lane halves
      p[nt] = t;
    }
    if (lane < 16) {
      float* o = out + (b0 + n) * 192 + layer * 64 + lane;
#pragma unroll
      for (int nt = 0; nt < 4; ++nt) o[nt * 16] = p[nt];
    }
  }
}

template <bool PACKED>
__global__ __launch_bounds__(WAVES_PER_BLOCK * 32)
void cross_net_kernel(const float* __restrict__ emb,   // (B, 32, 16)
                      const float* __restrict__ W0,    // layer-0 weights (maybe packed)
                      const float* __restrict__ W1,
                      const float* __restrict__ W2,
                      float* __restrict__ out) {       // (B, 192)
  __shared__ float s_x0[WAVES_PER_BLOCK][NB][16][X0_STRIDE];
  __shared__ float s_xk[WAVES_PER_BLOCK][NB][16][XK_STRIDE];

  const int lane = threadIdx.x & 31;
  const int wv = threadIdx.x >> 5;
  const int col = lane & 15;
  const int q = lane >> 4;
  const int kk = q * 2;
  const int koff = kk * 64 + col;
  const long b0 = ((long)blockIdx.x * WAVES_PER_BLOCK + wv) * NB;

  // ---- stage x0 tile into LDS as [d][m] (coalesced 128B global reads) ----
#pragma unroll
  for (int n = 0; n < NB; ++n) {
    const float* e = emb + (b0 + n) * 512;
#pragma unroll
    for (int r = 0; r < 16; ++r) {
      int idx = r * 32 + lane;  // idx = m*16 + d
      s_x0[wv][n][idx & 15][idx >> 4] = e[idx];
    }
  }
  __syncthreads();

  const float* s0 = &s_x0[wv][0][col][0];
  const float* s1 = &s_x0[wv][1][col][0];

  // ---- layer 0: xk == x0 (HK1 = 32) ----
  v2f x0f[NB][8];
#pragma unroll
  for (int n = 0; n < NB; ++n)
#pragma unroll
    for (int g = 0; g < 8; ++g)
      x0f[n][g] = *(const v2f*)&s_x0[wv][n][col][4 * g + kk];

  v8f acc[NB][4];
#pragma unroll
  for (int n = 0; n < NB; ++n)
#pragma unroll
    for (int nt = 0; nt < 4; ++nt)
#pragma unroll
      for (int j = 0; j < 8; ++j) acc[n][nt][j] = 0.0f;

  cross_layer<32, PACKED>(W0, s0, s1, x0f, acc, lane, koff);

  v2f xkf[NB][16];
#pragma unroll
  for (int L = 0; L < 2; ++L) {
    // relu
#pragma unroll
    for (int n = 0; n < NB; ++n)
#pragma unroll
      for (int nt = 0; nt < 4; ++nt)
#pragma unroll
        for (int j = 0; j < 8; ++j)
          acc[n][nt][j] = fmaxf(acc[n][nt][j], 0.0f);

    store_sums(acc, out, b0, L, lane);

    // C layout -> LDS [d][k]  (row d = j + 8*q, col k = nt*16+col)
#pragma unroll
    for (int n = 0; n < NB; ++n)
#pragma unroll
      for (int nt = 0; nt < 4; ++nt)
#pragma unroll
        for (int j = 0; j < 8; ++j)
          s_xk[wv][n][j + 8 * q][nt * 16 + col] = acc[n][nt][j];
    __syncthreads();

    // reload as A-fragments (ds_load_b64)
#pragma unroll
    for (int n = 0; n < NB; ++n)
#pragma unroll
      for (int g = 0; g < 16; ++g)
        xkf[n][g] = *(const v2f*)&s_xk[wv][n][col][4 * g + kk];
    __syncthreads();

#pragma unroll
    for (int n = 0; n < NB; ++n)
#pragma unroll
      for (int nt = 0; nt < 4; ++nt)
#pragma unroll
        for (int j = 0; j < 8; ++j) acc[n][nt][j] = 0.0f;

    cross_layer<64, PACKED>((L == 0) ? W1 : W2, s0, s1, xkf, acc, lane, koff);
  }

  // ---- finish layer 2 ----
#pragma unroll
  for (int n = 0; n < NB; ++n)
#pragma unroll
    for (int nt = 0; nt < 4; ++nt)
#pragma unroll
      for (int j = 0; j < 8; ++j)
        acc[n][nt][j] = fmaxf(acc[n][nt][j], 0.0f);

  store_sums(acc, out, b0, 2, lane);
}

// ---- one-shot weight repack into d_ws (fragment-order, b128 per lane) ----
__global__ __launch_bounds__(256)
void repack_w_kernel(const float* __restrict__ W0, const float* __restrict__ W1,
                     const float* __restrict__ W2, float* __restrict__ wp) {
  int t = blockIdx.x * blockDim.x + threadIdx.x;  // one v4 entry
  if (t >= PACK_V4_TOTAL) return;

  const float* W;
  int HK1, e;
  float* dst;
  if (t < L0_V4) {
    W = W0; HK1 = 32; e = t; dst = wp;
  } else if (t < L0_V4 + L1_V4) {
    W = W1; HK1 = 64; e = t - L0_V4; dst = wp + (size_t)L0_V4 * 4;
  } else {
    W = W2; HK1 = 64; e = t - (L0_V4 + L1_V4);
    dst = wp + (size_t)(L0_V4 + L1_V4) * 4;
  }

  int lane = e & 31;
  int np = (e >> 5) & 1;
  int rest = e >> 6;
  int g = rest % (HK1 / 4);
  int m = rest / (HK1 / 4);
  int col = lane & 15;
  int kk = (lane >> 4) * 2;
  int h = 4 * g + kk;
  int k0 = np * 32 + col;
  const float* src = W + (size_t)(m * HK1 + h) * 64;

  v4f v;
  v.x = src[k0];
  v.y = src[64 + k0];
  v.z = src[k0 + 16];
  v.w = src[64 + k0 + 16];
  *(v4f*)(dst + (size_t)e * 4) = v;
}

extern "C" void kernel_launch(void* const* d_in, const int* in_sizes, int n_in,
                              void* d_out, int out_size, void* d_ws, size_t ws_size,
                              hipStream_t stream) {
  const float* emb = (const float*)d_in[0];
  const float* W0 = (const float*)d_in[1];
  const float* W1 = (const float*)d_in[2];
  const float* W2 = (const float*)d_in[3];
  float* out = (float*)d_out;

  const int B = in_sizes[0] / 512;  // B*M*D / (32*16)
  const int batches_per_block = WAVES_PER_BLOCK * NB;
  dim3 grid((B + batches_per_block - 1) / batches_per_block);
  dim3 block(WAVES_PER_BLOCK * 32);

  const size_t pack_bytes = (size_t)PACK_V4_TOTAL * 16;
  if (ws_size >= pack_bytes) {
    float* wp = (float*)d_ws;
    hipLaunchKernelGGL(repack_w_kernel, dim3((PACK_V4_TOTAL + 255) / 256),
                       dim3(256), 0, stream, W0, W1, W2, wp);
    hipLaunchKernelGGL((cross_net_kernel<true>), grid, block, 0, stream, emb,
                       wp, wp + (size_t)L0_V4 * 4,
                       wp + (size_t)(L0_V4 + L1_V4) * 4, out);
  } else {
    hipLaunchKernelGGL((cross_net_kernel<false>), grid, block, 0, stream, emb,
                       W0, W1, W2, out);
  }
}